// VenomMaskedLinear_9878424780932
// MI455X (gfx1250) — compile-verified
//
#include <hip/hip_runtime.h>

typedef _Float16       half8   __attribute__((ext_vector_type(8)));
typedef _Float16       v16h    __attribute__((ext_vector_type(16)));
typedef _Float16       v32h    __attribute__((ext_vector_type(32)));
typedef float          v8f     __attribute__((ext_vector_type(8)));
typedef unsigned short ushort8 __attribute__((ext_vector_type(8)));

#define K_DIM 4096   // inner dim (weight cols / x cols)
#define N_DIM 4096   // out features (weight rows)
#define M_DIM 8192   // x rows
#define SK_TOT 1024  // stored (compressed) K per weight row  (2 per 8-group)
#define EK_TOT 2048  // expanded K (4 selected slots per 8-group)

// ---------------------------------------------------------------------------
// Kernel 0: x f32 -> f16 (one bandwidth pass; the gather then moves 2B/elem)
// ---------------------------------------------------------------------------
__global__ __launch_bounds__(256)
void cvt_x_f16(const float* __restrict__ X, _Float16* __restrict__ Xh)
{
    const size_t i = ((size_t)blockIdx.x * 256 + threadIdx.x) * 8;
    const float4 f0 = *reinterpret_cast<const float4*>(&X[i]);
    const float4 f1 = *reinterpret_cast<const float4*>(&X[i + 4]);
    half8 h;
    h[0] = (_Float16)f0.x; h[1] = (_Float16)f0.y;
    h[2] = (_Float16)f0.z; h[3] = (_Float16)f0.w;
    h[4] = (_Float16)f1.x; h[5] = (_Float16)f1.y;
    h[6] = (_Float16)f1.z; h[7] = (_Float16)f1.w;
    *reinterpret_cast<half8*>(&Xh[i]) = h;
}

// ---------------------------------------------------------------------------
// Kernel 1: VENOM select + 2:4 compress.
// Block = (ob: 64-row weight block) x (c32: 8 column groups = 64 cols).
// Emits: Wc   f16  [4096][1024]  compressed values (slot order, idx0<idx1)
//        Meta u32  [4096][64]    2-bit index pairs, nibble per group
//        SrcIdx u16 [64][2048]   expanded slot -> source column
// ---------------------------------------------------------------------------
__global__ __launch_bounds__(64)
void venom_select_compress(const float* __restrict__ W,
                           _Float16* __restrict__ Wc,
                           unsigned* __restrict__ Meta,
                           unsigned short* __restrict__ SrcIdx)
{
    const int c32 = blockIdx.x;   // 0..63
    const int ob  = blockIdx.y;   // 0..63
    const int t   = threadIdx.x;  // 0..63 = row within block

    __shared__ float         sW[64][65];   // +1 pad breaks bank conflicts
    __shared__ float         sScore[64];
    __shared__ unsigned char sTc[8][4];

    const int row  = ob * 64 + t;
    const int col0 = c32 * 64;

    #pragma unroll
    for (int c = 0; c < 64; c += 4) {
        const float4 f = *reinterpret_cast<const float4*>(
            &W[(size_t)row * K_DIM + col0 + c]);
        sW[t][c + 0] = f.x; sW[t][c + 1] = f.y;
        sW[t][c + 2] = f.z; sW[t][c + 3] = f.w;
    }
    __syncthreads();

    // per-column L1 mass over 64 rows (fixed order -> deterministic)
    {
        float s = 0.f;
        for (int r = 0; r < 64; ++r) s += fabsf(sW[r][t]);
        sScore[t] = s;
    }
    __syncthreads();

    // top-4 of 8 per group, score-descending, earlier index wins ties
    if (t < 8) {
        float sc[8];
        #pragma unroll
        for (int c = 0; c < 8; ++c) sc[c] = sScore[t * 8 + c];
        #pragma unroll
        for (int j = 0; j < 4; ++j) {
            int best = 0; float bv = sc[0];
            #pragma unroll
            for (int c = 1; c < 8; ++c)
                if (sc[c] > bv) { bv = sc[c]; best = c; }
            sTc[t][j] = (unsigned char)best;
            sc[best] = -3.402823466e38f;
        }
    }
    __syncthreads();

    // expanded-slot -> absolute source column map
    if (t < 32) {
        const int gl = t >> 2, j = t & 3;
        SrcIdx[(size_t)ob * EK_TOT + c32 * 32 + gl * 4 + j] =
            (unsigned short)(col0 + gl * 8 + sTc[gl][j]);
    }

    // per-row 2:4 among the 4 selected slots; pack values + 2-bit indices
    unsigned meta = 0;
    _Float16 packed[16];
    #pragma unroll
    for (int gl = 0; gl < 8; ++gl) {
        int   tc[4]; float vals[4];
        #pragma unroll
        for (int j = 0; j < 4; ++j) {
            tc[j]   = sTc[gl][j];
            vals[j] = fabsf(sW[t][gl * 8 + tc[j]]);
        }
        int i0 = 0; float v0 = vals[0];
        #pragma unroll
        for (int j = 1; j < 4; ++j) if (vals[j] > v0) { v0 = vals[j]; i0 = j; }
        int i1 = (i0 == 0) ? 1 : 0; float v1 = vals[i1];
        #pragma unroll
        for (int j = 0; j < 4; ++j)
            if (j != i0 && vals[j] > v1) { v1 = vals[j]; i1 = j; }
        const int jlo = (i0 < i1) ? i0 : i1;
        const int jhi = (i0 < i1) ? i1 : i0;
        packed[gl * 2 + 0] = (_Float16)sW[t][gl * 8 + tc[jlo]];
        packed[gl * 2 + 1] = (_Float16)sW[t][gl * 8 + tc[jhi]];
        meta |= (unsigned)(jlo | (jhi << 2)) << (gl * 4);
    }
    Meta[(size_t)row * 64 + c32] = meta;

    half8 p0, p1;
    #pragma unroll
    for (int j = 0; j < 8; ++j) { p0[j] = packed[j]; p1[j] = packed[j + 8]; }
    _Float16* wcp = &Wc[(size_t)row * SK_TOT + c32 * 16];
    *reinterpret_cast<half8*>(wcp)     = p0;
    *reinterpret_cast<half8*>(wcp + 8) = p1;
}

// ---------------------------------------------------------------------------
// Kernel 2: sparse GEMM via v_swmmac_f32_16x16x64_f16.
// D[o][m] (=y^T) : A = compressed weights (sparse M=o), B = gathered x.
// Block: one ob (64 o-rows) x 256 m-rows; 8 waves = 2(o) x 4(m);
// wave = 2 o-frags x 4 m-frags. 32 stages of EK=64 (16 groups / 128 src cols).
// ---------------------------------------------------------------------------
#define BO   64
#define BMX  256
#define LDBG 72      // 64 + 8 halfs: 144B row stride, 16B aligned

__global__ __launch_bounds__(256)
void venom_spmm(const _Float16* __restrict__ Xh,
                const _Float16* __restrict__ Wc,
                const unsigned* __restrict__ Meta,
                const unsigned short* __restrict__ SrcIdx,
                const float* __restrict__ bias,
                float* __restrict__ Y)
{
    __shared__ __align__(16) _Float16 Bg[BMX * LDBG];

    const int tid  = threadIdx.x;
    const int lane = tid & 31;
    const int wave = tid >> 5;
    const int waveO  = wave >> 2;   // 0..1 -> 32 o-rows
    const int waveMx = wave & 3;    // 0..3 -> 64 m-rows
    const int lr = lane & 15;
    const int lh = lane >> 4;

    const int ob  = blockIdx.y;
    const int ob0 = ob * BO;
    const int m0  = blockIdx.x * BMX;

    const unsigned short* sidx = &SrcIdx[(size_t)ob * EK_TOT];

    v8f acc[2][4];
    #pragma unroll
    for (int f = 0; f < 2; ++f)
        #pragma unroll
        for (int n = 0; n < 4; ++n)
            #pragma unroll
            for (int j = 0; j < 8; ++j)
                acc[f][n][j] = 0.f;

    const int oct   = tid & 7;    // 8-half octet within ek
    const int mloc0 = tid >> 3;   // 0..31: m row within pass

    for (int es = 0; es < EK_TOT / 64; ++es) {
        // ---- gather-stage x into Bg[m][ek] ----
        const ushort8 idx8 = *reinterpret_cast<const ushort8*>(
            &sidx[es * 64 + oct * 8]);
        #pragma unroll
        for (int pass = 0; pass < 8; ++pass) {
            const int ml = pass * 32 + mloc0;
            const size_t xrow = (size_t)(m0 + ml) * K_DIM;
            half8 h;
            #pragma unroll
            for (int j = 0; j < 8; ++j)
                h[j] = Xh[xrow + idx8[j]];
            *reinterpret_cast<half8*>(&Bg[ml * LDBG + oct * 8]) = h;
        }
        __syncthreads();

        // ---- A (sparse 16x32 stored) + index metadata, straight from L2 ----
        v16h a[2]; int am[2];
        #pragma unroll
        for (int f = 0; f < 2; ++f) {
            const int o = ob0 + waveO * 32 + f * 16 + lr;
            const _Float16* p = &Wc[(size_t)o * SK_TOT + es * 32 + lh * 8];
            const half8 lo = *reinterpret_cast<const half8*>(p);
            const half8 hi = *reinterpret_cast<const half8*>(p + 16);
            a[f] = __builtin_shufflevector(lo, hi,
                0,1,2,3,4,5,6,7,8,9,10,11,12,13,14,15);
            am[f] = (int)Meta[(size_t)o * 64 + es * 2 + lh];
        }

        // ---- B frags (64x16): lane holds ek chunks {lh*16, 32+lh*16} ----
        v32h b[4];
        #pragma unroll
        for (int n = 0; n < 4; ++n) {
            const int mm = waveMx * 64 + n * 16 + lr;
            const _Float16* p = &Bg[mm * LDBG + lh * 16];
            const half8 q0 = *reinterpret_cast<const half8*>(p);
            const half8 q1 = *reinterpret_cast<const half8*>(p + 8);
            const half8 q2 = *reinterpret_cast<const half8*>(p + 32);
            const half8 q3 = *reinterpret_cast<const half8*>(p + 40);
            const v16h lo16 = __builtin_shufflevector(q0, q1,
                0,1,2,3,4,5,6,7,8,9,10,11,12,13,14,15);
            const v16h hi16 = __builtin_shufflevector(q2, q3,
                0,1,2,3,4,5,6,7,8,9,10,11,12,13,14,15);
            b[n] = __builtin_shufflevector(lo16, hi16,
                0,1,2,3,4,5,6,7,8,9,10,11,12,13,14,15,
                16,17,18,19,20,21,22,23,24,25,26,27,28,29,30,31);
        }

        #pragma unroll
        for (int f = 0; f < 2; ++f)
            #pragma unroll
            for (int n = 0; n < 4; ++n)
                acc[f][n] = __builtin_amdgcn_swmmac_f32_16x16x64_f16(
                    false, a[f], false, b[n], acc[f][n], am[f], false, false);
        __syncthreads();
    }

    // ---- epilogue: D vgpr j -> o = base+j(+8*lh); lane lr -> m. y[m][o]+b ----
    typedef float float8 __attribute__((ext_vector_type(8)));
    #pragma unroll
    for (int f = 0; f < 2; ++f) {
        const int obase = ob0 + waveO * 32 + f * 16 + lh * 8;
        const float8 bv = *reinterpret_cast<const float8*>(&bias[obase]);
        #pragma unroll
        for (int n = 0; n < 4; ++n) {
            const int m = m0 + waveMx * 64 + n * 16 + lr;
            float8 r;
            #pragma unroll
            for (int j = 0; j < 8; ++j) r[j] = acc[f][n][j] + bv[j];
            *reinterpret_cast<float8*>(&Y[(size_t)m * N_DIM + obase]) = r;
        }
    }
}

// ---------------------------------------------------------------------------
extern "C" void kernel_launch(void* const* d_in, const int* in_sizes, int n_in,
                              void* d_out, int out_size, void* d_ws, size_t ws_size,
                              hipStream_t stream)
{
    (void)in_sizes; (void)n_in; (void)out_size; (void)ws_size;

    const float* x    = (const float*)d_in[0];
    const float* w    = (const float*)d_in[1];
    const float* bias = (const float*)d_in[2];
    float*       y    = (float*)d_out;

    // workspace layout
    char* ws = (char*)d_ws;
    _Float16*       xh   = (_Float16*)ws;                               // 64 MB
    _Float16*       wc   = (_Float16*)(ws + (size_t)64 * 1024 * 1024);  //  8 MB
    unsigned*       meta = (unsigned*)(ws + (size_t)72 * 1024 * 1024);  //  1 MB
    unsigned short* sidx = (unsigned short*)(ws + (size_t)73 * 1024 * 1024); // 256 KB

    cvt_x_f16<<<(M_DIM * K_DIM) / (256 * 8), 256, 0, stream>>>(x, xh);

    dim3 g1(K_DIM / 64, N_DIM / 64);   // (c32=64, ob=64)
    venom_select_compress<<<g1, 64, 0, stream>>>(w, wc, meta, sidx);

    dim3 g2(M_DIM / BMX, N_DIM / BO);  // (32, 64)
    venom_spmm<<<g2, 256, 0, stream>>>(xh, wc, meta, sidx, bias, y);
}